// HierarchicalMemory_67997922230380
// MI455X (gfx1250) — compile-verified
//
#include <hip/hip_runtime.h>
#include <math.h>

typedef __attribute__((ext_vector_type(2))) float v2f;
typedef __attribute__((ext_vector_type(8))) float v8f;

#define EPSV 1e-8f
#define HDIM 256

// ---------------------------------------------------------------------------
// Kernel 1: normalize query rows: qn = q / max(||q||, eps). One wave per row.
// ---------------------------------------------------------------------------
__global__ __launch_bounds__(256) void qnorm_kernel(const float* __restrict__ q,
                                                    float* __restrict__ qn, int B) {
  int row  = blockIdx.x * 8 + (threadIdx.x >> 5);
  int lane = threadIdx.x & 31;
  if (row >= B) return;
  const float* p = q + (size_t)row * HDIM + lane * 8;
  float4 a = *(const float4*)p;
  float4 b = *(const float4*)(p + 4);
  float s = a.x*a.x + a.y*a.y + a.z*a.z + a.w*a.w
          + b.x*b.x + b.y*b.y + b.z*b.z + b.w*b.w;
  #pragma unroll
  for (int off = 16; off > 0; off >>= 1) s += __shfl_xor(s, off, 32);
  float inv = 1.0f / fmaxf(sqrtf(s), EPSV);
  float* o = qn + (size_t)row * HDIM + lane * 8;
  float4 oa, ob;
  oa.x = a.x*inv; oa.y = a.y*inv; oa.z = a.z*inv; oa.w = a.w*inv;
  ob.x = b.x*inv; ob.y = b.y*inv; ob.z = b.z*inv; ob.w = b.w*inv;
  *(float4*)o       = oa;
  *(float4*)(o + 4) = ob;
}

// ---------------------------------------------------------------------------
// Kernel 2: sims = qn @ emb^T / max(||emb_row||, eps) via V_WMMA_F32_16X16X4_F32.
// One wave per 16-column block of embeddings, computing ALL MB m-bands (B<=64)
// so each B fragment (the bandwidth-dominant operand) is loaded exactly once.
// A frag (16x4 f32): M = lane%16, K = 2*(lane/16) + vgpr  -> float2 load.
// B frag (4x16 f32): N = lane%16, K = 2*(lane/16) + vgpr  -> float2 load.
// C/D (16x16 f32):   N = lane%16, M = vgpr + 8*(lane/16).
// ---------------------------------------------------------------------------
__global__ __launch_bounds__(256) void sim_kernel(const float* __restrict__ qn,
                                                  const float* __restrict__ emb,
                                                  float* __restrict__ sims,
                                                  int C, int ntiles, int MB) {
  int wid  = threadIdx.x >> 5;
  int lane = threadIdx.x & 31;
  int tile = blockIdx.x * 8 + wid;
  if (tile >= ntiles) return;              // wave-uniform
  int n0 = tile * 16;
  int r  = lane & 15;
  int hl = lane >> 4;

  const float* ap = qn  + (size_t)r * HDIM + 2 * hl;       // + mb*16*HDIM + k0
  const float* bp = emb + (size_t)(n0 + r) * HDIM + 2 * hl;

  v8f acc[4];
  #pragma unroll
  for (int mb = 0; mb < 4; mb++) acc[mb] = (v8f){};

  float esq = 0.0f;
  #pragma unroll 4
  for (int k0 = 0; k0 < HDIM; k0 += 4) {
    v2f bb = *(const v2f*)(bp + k0);
    esq += bb.x * bb.x + bb.y * bb.y;
    #pragma unroll
    for (int mb = 0; mb < 4; mb++) {
      v2f a = *(const v2f*)(ap + mb * 16 * HDIM + k0);
      acc[mb] = __builtin_amdgcn_wmma_f32_16x16x4_f32(
          /*neg_a=*/false, a, /*neg_b=*/false, bb,
          /*c_mod=*/(short)0, acc[mb], /*reuse_a=*/false, /*reuse_b=*/false);
    }
  }
  // lane and lane^16 together cover the full K range of embedding row n0+r
  float efull = esq + __shfl_xor(esq, 16, 32);
  float scale = 1.0f / fmaxf(sqrtf(efull), EPSV);

  int n = n0 + r;
  #pragma unroll
  for (int mb = 0; mb < 4; mb++) {
    if (mb >= MB) break;                   // uniform (MB==4 for this problem)
    #pragma unroll
    for (int v = 0; v < 8; v++) {
      int m = mb * 16 + v + 8 * hl;
      sims[(size_t)m * C + n] = acc[mb][v] * scale;
    }
  }
}

// ---------------------------------------------------------------------------
// Kernel 3: top-k (k<=8) per row. One wave per row; lane-local sorted top-8,
// then k rounds of wave argmax (tie-break: lower index).
// ---------------------------------------------------------------------------
__device__ __forceinline__ bool betterf(float s1, int i1, float s2, int i2) {
  return (s1 > s2) || (s1 == s2 && i1 < i2);
}

__global__ __launch_bounds__(32) void topk_kernel(const float* __restrict__ sims,
                                                  float* __restrict__ out_scores,
                                                  int* __restrict__ out_idx,
                                                  int C, int k) {
  int b    = blockIdx.x;
  int lane = threadIdx.x;
  const float* row = sims + (size_t)b * C;

  float sc[8]; int id[8];
  #pragma unroll
  for (int i = 0; i < 8; i++) { sc[i] = -INFINITY; id[i] = 0x7fffffff; }

  for (int c = lane; c < C; c += 32) {
    float s = row[c];
    if (betterf(s, c, sc[7], id[7])) {
      int j = 7;
      while (j > 0 && betterf(s, c, sc[j - 1], id[j - 1])) {
        sc[j] = sc[j - 1]; id[j] = id[j - 1]; --j;
      }
      sc[j] = s; id[j] = c;
    }
  }

  int p = 0;
  for (int r = 0; r < k; r++) {
    float bs = (p < 8) ? sc[p] : -INFINITY;
    int   bi = (p < 8) ? id[p] : 0x7fffffff;
    #pragma unroll
    for (int off = 16; off > 0; off >>= 1) {
      float s2 = __shfl_xor(bs, off, 32);
      int   i2 = __shfl_xor(bi, off, 32);
      if (betterf(s2, i2, bs, bi)) { bs = s2; bi = i2; }
    }
    if (p < 8 && sc[p] == bs && id[p] == bi) p++;   // unique winner advances
    if (lane == 0) {
      out_scores[b * k + r] = bs;
      out_idx[b * k + r]    = bi;
    }
  }
}

// ---------------------------------------------------------------------------
// Kernel 4: gather retrieved[b,j,s,:] from episodes / zero-padded compressed.
// One float4 per thread.
// ---------------------------------------------------------------------------
__global__ __launch_bounds__(256) void gather_kernel(const float* __restrict__ episodes,
                                                     const float* __restrict__ comp,
                                                     const int* __restrict__ is_comp,
                                                     const int* __restrict__ top_idx,
                                                     float* __restrict__ out,
                                                     int total4, int S, int Cs, int H4) {
  int t = blockIdx.x * 256 + threadIdx.x;
  if (t >= total4) return;
  int h4 = t % H4;
  int r2 = t / H4;
  int s  = r2 % S;
  int bj = r2 / S;
  int idx = top_idx[bj];

  float4 v; v.x = 0.0f; v.y = 0.0f; v.z = 0.0f; v.w = 0.0f;
  if (is_comp[idx] != 0) {
    if (s < Cs)
      v = ((const float4*)comp)[((size_t)idx * Cs + s) * H4 + h4];
  } else {
    v = ((const float4*)episodes)[((size_t)idx * S + s) * H4 + h4];
  }
  ((float4*)out)[t] = v;
}

// ---------------------------------------------------------------------------
extern "C" void kernel_launch(void* const* d_in, const int* in_sizes, int n_in,
                              void* d_out, int out_size, void* d_ws, size_t ws_size,
                              hipStream_t stream) {
  const float* query    = (const float*)d_in[0];
  const float* episodes = (const float*)d_in[1];
  const float* comp     = (const float*)d_in[2];
  const float* emb      = (const float*)d_in[3];
  const int*   is_comp  = (const int*)d_in[4];

  const int Hc = HDIM;
  int B  = in_sizes[0] / Hc;                 // 64
  int C  = in_sizes[3] / Hc;                 // 20000
  int S  = in_sizes[1] / (C * Hc);           // 32
  int Cs = in_sizes[2] / (C * Hc);           // 16
  int k  = out_size / (B * (S * Hc + 1));    // 8 (retrieved + scores, flat)

  // Workspace layout
  float* qn      = (float*)d_ws;                      // B*H floats
  float* sims    = qn + (size_t)B * Hc;               // B*C floats
  int*   top_idx = (int*)(sims + (size_t)B * C);      // B*k ints

  float* out_retr   = (float*)d_out;
  float* out_scores = out_retr + (size_t)B * k * S * Hc;

  // 1) normalize queries (wave per row)
  qnorm_kernel<<<(B + 7) / 8, 256, 0, stream>>>(query, qn, B);

  // 2) WMMA similarity GEMM: one wave per 16-col block, all m-bands resident
  int MB     = B / 16;                       // 4
  int ntiles = C / 16;                       // 1250
  sim_kernel<<<(ntiles + 7) / 8, 256, 0, stream>>>(qn, emb, sims, C, ntiles, MB);

  // 3) top-k per row (wave per row); scores go straight to output tail
  topk_kernel<<<B, 32, 0, stream>>>(sims, out_scores, top_idx, C, k);

  // 4) gather selected episodes into output
  int total4 = B * k * S * (Hc / 4);
  gather_kernel<<<(total4 + 255) / 256, 256, 0, stream>>>(
      episodes, comp, is_comp, top_idx, out_retr, total4, S, Cs, Hc / 4);
}